// GATEncoder_42820823941371
// MI455X (gfx1250) — compile-verified
//
#include <hip/hip_runtime.h>
#include <math.h>

// Problem constants (match reference)
constexpr int N_NODES  = 50000;
constexpr int N_EDGES  = 800000;
constexpr int N_GRAPHS = 64;
constexpr int IN_FEATS = 128;
constexpr int HID      = 64;
constexpr int HEADS    = 4;
constexpr int HF       = HEADS * HID;   // 256
#define LEAKY 0.2f

typedef __attribute__((ext_vector_type(2))) float v2f;
typedef __attribute__((ext_vector_type(8))) float v8f;

// ---- monotone float<->int encoding for integer atomicMax on floats --------
__device__ __forceinline__ int enc_f32(float f) {
    int i = __float_as_int(f);
    return i ^ ((i >> 31) & 0x7FFFFFFF);   // involution, order-preserving
}
__device__ __forceinline__ float dec_f32(int i) {
    int j = i ^ ((i >> 31) & 0x7FFFFFFF);
    return __int_as_float(j);
}

// ---- utility fills --------------------------------------------------------
__global__ void zero_f32_kernel(float* p, long n) {
    long i = (long)blockIdx.x * blockDim.x + threadIdx.x;
    long st = (long)gridDim.x * blockDim.x;
    for (; i < n; i += st) p[i] = 0.0f;
}
__global__ void fill_i32_kernel(int* p, int v, long n) {
    long i = (long)blockIdx.x * blockDim.x + threadIdx.x;
    long st = (long)gridDim.x * blockDim.x;
    for (; i < n; i += st) p[i] = v;
}

// ---- fp32 GEMM via V_WMMA_F32_16X16X4_F32 ---------------------------------
// C[M,Ncol] = A[M,K] * B[K,Ncol], row-major. M%16==0, K%4==0, Ncol%64==0.
// grid: (M/16, Ncol/64), block: 128 (4 waves); wave w -> 16x16 tile.
__global__ void gemm_f32_wmma(const float* __restrict__ A,
                              const float* __restrict__ B,
                              float* __restrict__ C,
                              int M, int K, int Ncol) {
    const int lane = threadIdx.x & 31;
    const int wave = threadIdx.x >> 5;
    const int m    = lane & 15;
    const int half = lane >> 4;
    const int row0 = blockIdx.x << 4;
    const int col0 = blockIdx.y * 64 + wave * 16;

    v8f acc = {};
    // A fragment: lane reads A[row0+m][k0 + 2*half + {0,1}]
    const float* arow = A + (size_t)(row0 + m) * K + 2 * half;
    // B fragment: lane reads B[k0 + 2*half + {0,1}][col0+m]
    const float* bcol = B + col0 + m;

    for (int k0 = 0; k0 < K; k0 += 4) {
        v2f a = *(const v2f*)(arow + k0);            // 8B-aligned (K%4==0)
        int kb = k0 + 2 * half;
        v2f b;
        b.x = bcol[(size_t)kb * Ncol];
        b.y = bcol[(size_t)(kb + 1) * Ncol];
        acc = __builtin_amdgcn_wmma_f32_16x16x4_f32(
            /*neg_a=*/false, a, /*neg_b=*/false, b,
            /*c_mod=*/(short)0, acc, /*reuse_a=*/false, /*reuse_b=*/false);
    }
    // C/D layout: VGPR r -> (lanes0-15: M=r, lanes16-31: M=8+r), N = lane%16
#pragma unroll
    for (int r = 0; r < 8; ++r) {
        int row = row0 + 8 * half + r;
        C[(size_t)row * Ncol + col0 + m] = acc[r];
    }
}

// ---- per-node attention scores: el/er[n,h] = sum_f h[n,h,f]*a{l,r}[h,f] ---
__global__ void scores_kernel(const float* __restrict__ h,
                              const float* __restrict__ al,
                              const float* __restrict__ ar,
                              float* __restrict__ el, float* __restrict__ er) {
    long i = (long)blockIdx.x * blockDim.x + threadIdx.x;
    if (i >= (long)N_NODES * HEADS) return;
    int n = (int)(i >> 2);
    int hh = (int)(i & 3);
    const float* hp  = h  + (size_t)n * HF + hh * HID;
    const float* alp = al + hh * HID;
    const float* arp = ar + hh * HID;
    float sl = 0.f, sr = 0.f;
#pragma unroll 8
    for (int f = 0; f < HID; ++f) { sl += hp[f] * alp[f]; sr += hp[f] * arp[f]; }
    el[i] = sl;
    er[i] = sr;
}

// ---- edge pass A: e = leaky(el[src]+er[dst]); segment max over dst --------
__global__ void edge_max_kernel(const int* __restrict__ src, const int* __restrict__ dst,
                                const float* __restrict__ el, const float* __restrict__ er,
                                float* __restrict__ e_out, int* __restrict__ menc) {
    long i = (long)blockIdx.x * blockDim.x + threadIdx.x;
    if (i >= (long)N_EDGES * HEADS) return;
    int e = (int)(i >> 2), hh = (int)(i & 3);
    int sn = src[e], dn = dst[e];
    float v = el[sn * HEADS + hh] + er[dn * HEADS + hh];
    v = (v >= 0.f) ? v : LEAKY * v;
    e_out[i] = v;
    atomicMax(&menc[dn * HEADS + hh], enc_f32(v));
}

// ---- edge pass B: ex = exp(e - m[dst]); segment sum over dst --------------
__global__ void edge_exp_kernel(const int* __restrict__ dst,
                                float* __restrict__ e_io,
                                const int* __restrict__ menc,
                                float* __restrict__ s) {
    long i = (long)blockIdx.x * blockDim.x + threadIdx.x;
    if (i >= (long)N_EDGES * HEADS) return;
    int e = (int)(i >> 2), hh = (int)(i & 3);
    int dn = dst[e];
    float mx = dec_f32(menc[dn * HEADS + hh]);
    float ex = expf(e_io[i] - mx);
    e_io[i] = ex;
    atomicAdd(&s[dn * HEADS + hh], ex);
}

// ---- edge pass C: acc[dst] += h[src]*alpha ; one block (256 thr) per edge -
__global__ void edge_agg_kernel(const int* __restrict__ src, const int* __restrict__ dst,
                                const float* __restrict__ hfeat,
                                const float* __restrict__ ex, const float* __restrict__ s,
                                float* __restrict__ acc) {
    int e = blockIdx.x;
    int t = threadIdx.x;           // 0..255 -> (h,f)
    int hh = t >> 6;
    int sn = src[e], dn = dst[e];
    float d = s[dn * HEADS + hh];
    d = (d == 0.f) ? 1.f : d;
    float alpha = ex[(size_t)e * HEADS + hh] / d;
    atomicAdd(&acc[(size_t)dn * HF + t], hfeat[(size_t)sn * HF + t] * alpha);
}

// ---- layer1 epilogue: x = elu(acc + b) in place ---------------------------
__global__ void bias_elu_kernel(float* __restrict__ p, const float* __restrict__ b) {
    long i = (long)blockIdx.x * blockDim.x + threadIdx.x;
    long st = (long)gridDim.x * blockDim.x;
    long n = (long)N_NODES * HF;
    for (; i < n; i += st) {
        float x = p[i] + b[i % HF];
        p[i] = (x > 0.f) ? x : expm1f(x);
    }
}

// ---- layer2 epilogue: hm[n,f] = mean_h(acc[n,h,f] + b2[h,f]) --------------
__global__ void mean_bias_kernel(const float* __restrict__ acc, const float* __restrict__ b2,
                                 float* __restrict__ hm) {
    long i = (long)blockIdx.x * blockDim.x + threadIdx.x;
    if (i >= (long)N_NODES * HID) return;
    int n = (int)(i / HID);
    int f = (int)(i % HID);
    float a = 0.f;
#pragma unroll
    for (int hh = 0; hh < HEADS; ++hh)
        a += acc[(size_t)n * HF + hh * HID + f] + b2[hh * HID + f];
    hm[i] = a * 0.25f;
}

// ---- readout gate: w[n] = sigmoid(hm[n,:]·Wr + br) ------------------------
__global__ void gate_kernel(const float* __restrict__ hm, const float* __restrict__ Wr,
                            const float* __restrict__ br, float* __restrict__ w) {
    int n = blockIdx.x * blockDim.x + threadIdx.x;
    if (n >= N_NODES) return;
    float a = 0.f;
#pragma unroll 8
    for (int f = 0; f < HID; ++f) a += hm[(size_t)n * HID + f] * Wr[f];
    a += br[0];
    w[n] = 1.f / (1.f + expf(-a));
}

// ---- readout: weighted segment sum (into out) + segment max (encoded) -----
__global__ void readout_kernel(const float* __restrict__ hm, const float* __restrict__ w,
                               const int* __restrict__ gid, float* __restrict__ out,
                               int* __restrict__ mxenc) {
    long i = (long)blockIdx.x * blockDim.x + threadIdx.x;
    if (i >= (long)N_NODES * HID) return;
    int n = (int)(i / HID);
    int f = (int)(i % HID);
    int g = gid[n];
    float v = hm[i];
    atomicAdd(&out[(size_t)g * (2 * HID) + f], v * w[n]);
    atomicMax(&mxenc[g * HID + f], enc_f32(v));
}

__global__ void finalize_max_kernel(const int* __restrict__ mxenc, float* __restrict__ out) {
    int i = blockIdx.x * blockDim.x + threadIdx.x;
    if (i >= N_GRAPHS * HID) return;
    int g = i / HID, f = i % HID;
    out[(size_t)g * (2 * HID) + HID + f] = dec_f32(mxenc[i]);
}

static inline int gblocks(long n, int b) { return (int)((n + b - 1) / b); }

extern "C" void kernel_launch(void* const* d_in, const int* in_sizes, int n_in,
                              void* d_out, int out_size, void* d_ws, size_t ws_size,
                              hipStream_t stream) {
    const float* feats = (const float*)d_in[0];
    const int*   src   = (const int*)d_in[1];
    const int*   dst   = (const int*)d_in[2];
    const int*   gid   = (const int*)d_in[3];
    const float* W1    = (const float*)d_in[4];
    const float* al1   = (const float*)d_in[5];
    const float* ar1   = (const float*)d_in[6];
    const float* b1    = (const float*)d_in[7];
    const float* W2    = (const float*)d_in[8];
    const float* al2   = (const float*)d_in[9];
    const float* ar2   = (const float*)d_in[10];
    const float* b2    = (const float*)d_in[11];
    const float* Wr    = (const float*)d_in[12];
    const float* br    = (const float*)d_in[13];
    float* out = (float*)d_out;

    // workspace layout
    char* ws = (char*)d_ws;
    float* bufA = (float*)ws;  ws += (size_t)N_NODES * HF * sizeof(float);      // h1 / h2
    float* bufB = (float*)ws;  ws += (size_t)N_NODES * HF * sizeof(float);      // acc1/x2/acc2
    float* esc  = (float*)ws;  ws += (size_t)N_EDGES * HEADS * sizeof(float);   // e/ex, later hm
    float* el   = (float*)ws;  ws += (size_t)N_NODES * HEADS * sizeof(float);
    float* er   = (float*)ws;  ws += (size_t)N_NODES * HEADS * sizeof(float);
    int*   menc = (int*)ws;    ws += (size_t)N_NODES * HEADS * sizeof(int);
    float* ssum = (float*)ws;  ws += (size_t)N_NODES * HEADS * sizeof(float);
    float* wbuf = (float*)ws;  ws += (size_t)N_NODES * sizeof(float);
    int*   mxen = (int*)ws;    ws += (size_t)N_GRAPHS * HID * sizeof(int);
    float* hm   = esc;         // reuse: N_NODES*HID <= N_EDGES*HEADS

    const long NH  = (long)N_NODES * HEADS;
    const long EH  = (long)N_EDGES * HEADS;
    const long NHF = (long)N_NODES * HF;

    dim3 g1(N_NODES / 16, HF / 64), b128(128);

    // ---------------- Layer 1 ----------------
    gemm_f32_wmma<<<g1, b128, 0, stream>>>(feats, W1, bufA, N_NODES, IN_FEATS, HF);
    scores_kernel<<<gblocks(NH, 256), 256, 0, stream>>>(bufA, al1, ar1, el, er);
    fill_i32_kernel<<<512, 256, 0, stream>>>(menc, (int)0x80000000, NH);
    zero_f32_kernel<<<512, 256, 0, stream>>>(ssum, NH);
    edge_max_kernel<<<gblocks(EH, 256), 256, 0, stream>>>(src, dst, el, er, esc, menc);
    edge_exp_kernel<<<gblocks(EH, 256), 256, 0, stream>>>(dst, esc, menc, ssum);
    zero_f32_kernel<<<2048, 256, 0, stream>>>(bufB, NHF);
    edge_agg_kernel<<<N_EDGES, HF, 0, stream>>>(src, dst, bufA, esc, ssum, bufB);
    bias_elu_kernel<<<2048, 256, 0, stream>>>(bufB, b1);

    // ---------------- Layer 2 ----------------
    gemm_f32_wmma<<<g1, b128, 0, stream>>>(bufB, W2, bufA, N_NODES, HF, HF);
    scores_kernel<<<gblocks(NH, 256), 256, 0, stream>>>(bufA, al2, ar2, el, er);
    fill_i32_kernel<<<512, 256, 0, stream>>>(menc, (int)0x80000000, NH);
    zero_f32_kernel<<<512, 256, 0, stream>>>(ssum, NH);
    edge_max_kernel<<<gblocks(EH, 256), 256, 0, stream>>>(src, dst, el, er, esc, menc);
    edge_exp_kernel<<<gblocks(EH, 256), 256, 0, stream>>>(dst, esc, menc, ssum);
    zero_f32_kernel<<<2048, 256, 0, stream>>>(bufB, NHF);   // bufB free after gemm2
    edge_agg_kernel<<<N_EDGES, HF, 0, stream>>>(src, dst, bufA, esc, ssum, bufB);
    mean_bias_kernel<<<gblocks((long)N_NODES * HID, 256), 256, 0, stream>>>(bufB, b2, hm);

    // ---------------- Readout ----------------
    gate_kernel<<<gblocks(N_NODES, 256), 256, 0, stream>>>(hm, Wr, br, wbuf);
    zero_f32_kernel<<<32, 256, 0, stream>>>(out, (long)N_GRAPHS * 2 * HID);
    fill_i32_kernel<<<32, 256, 0, stream>>>(mxen, (int)0x80000000, (long)N_GRAPHS * HID);
    readout_kernel<<<gblocks((long)N_NODES * HID, 256), 256, 0, stream>>>(hm, wbuf, gid, out, mxen);
    finalize_max_kernel<<<gblocks((long)N_GRAPHS * HID, 256), 256, 0, stream>>>(mxen, out);
}